// GCPLLoss_37271726194988
// MI455X (gfx1250) — compile-verified
//
#include <hip/hip_runtime.h>

#define C_CLS 10
#define P_CAP 128
#define D_DIM 512
#define B_N   8192
#define CP    (C_CLS * P_CAP)   /* 1280 */
#define EPSV  1e-6f
#define GAMMAV 0.1f

#define LDS_STRIDE 36            /* 32 + 4 pad: float4-aligned, bank-friendly */

typedef __attribute__((ext_vector_type(2))) float v2f;
typedef __attribute__((ext_vector_type(8))) float v8f;

// ---------------------------------------------------------------------------
// Kernel 0: initialize workspace + output
// ---------------------------------------------------------------------------
__global__ void init_ws(float* __restrict__ protos, float* __restrict__ pcs,
                        unsigned* __restrict__ dminbits, int* __restrict__ nproto,
                        float* __restrict__ out) {
    int i = blockIdx.x * blockDim.x + threadIdx.x;
    if (i < CP * D_DIM) protos[i] = 0.0f;
    if (i < B_N * C_CLS) { pcs[i] = 0.0f; dminbits[i] = 0x7F800000u; /* +inf */ }
    if (i < C_CLS) nproto[i] = 0;
    if (i == 0) out[0] = 0.0f;
}

// ---------------------------------------------------------------------------
// Kernel 1: sequential prototype assignment (single workgroup, 8 waves)
// counts/nproto live in LDS for the whole scan; protos in global workspace.
// ---------------------------------------------------------------------------
__global__ void assign_protos(const float* __restrict__ F, const int* __restrict__ L,
                              float* __restrict__ protos, int* __restrict__ nproto_g) {
    __shared__ float sfeat[D_DIM];
    __shared__ float sd2[P_CAP];
    __shared__ int   s_counts[CP];
    __shared__ int   s_n[C_CLS];
    __shared__ int   s_tgt, s_do;
    __shared__ float s_cold;

    const int tid  = threadIdx.x;           // 0..255
    const int w    = tid >> 5;              // wave id 0..7
    const int lane = tid & 31;

    for (int i = tid; i < CP; i += 256) s_counts[i] = 0;
    if (tid < C_CLS) s_n[tid] = 0;
    __syncthreads();

    for (int i = 0; i < B_N; ++i) {
        const int l = L[i];
        const float* frow = F + (long long)i * D_DIM;
        sfeat[tid]       = frow[tid];
        sfeat[tid + 256] = frow[tid + 256];
        __syncthreads();

        const int n = s_n[l];
        const float* pbase = protos + (long long)l * P_CAP * D_DIM;

        // 8 waves x 16 slots each = 128 prototype slots; lane-strided coalesced reads
        for (int s = 0; s < 16; ++s) {
            const int p = w + 8 * s;
            float partial;
            if (p < n) {
                const float* prow = pbase + (long long)p * D_DIM;
                partial = 0.0f;
                for (int j = 0; j < 16; ++j) {
                    const int d = lane + 32 * j;
                    const float diff = sfeat[d] - prow[d] + EPSV;
                    partial += diff * diff;
                }
                for (int off = 16; off > 0; off >>= 1)
                    partial += __shfl_xor(partial, off, 32);
            } else {
                partial = __builtin_inff();
            }
            if (lane == 0) sd2[p] = partial;
        }
        __syncthreads();

        if (tid == 0) {
            int idx = 0; float mn = __builtin_inff();
            for (int p = 0; p < n; ++p) { const float v = sd2[p]; if (v < mn) { mn = v; idx = p; } }
            const float dmin = sqrtf(mn);
            const int doup = (n > 0) && (dmin < 40.0f);
            const int app  = (n < P_CAP - 1) ? n : (P_CAP - 1);
            const int tgt  = doup ? idx : app;
            const int cold = s_counts[l * P_CAP + idx];
            s_cold = (float)cold;
            s_counts[l * P_CAP + tgt] = doup ? (cold + 1) : 1;
            s_n[l] = doup ? n : ((n + 1 < P_CAP) ? (n + 1) : P_CAP);
            s_tgt = tgt; s_do = doup;
        }
        __syncthreads();

        {
            float* trow = protos + (long long)l * P_CAP * D_DIM + (long long)s_tgt * D_DIM;
            const float c = s_cold;
            const int doup = s_do;
            for (int d = tid; d < D_DIM; d += 256) {
                const float old = trow[d];
                trow[d] = doup ? (old * c + sfeat[d]) / (c + 1.0f) : sfeat[d];
            }
        }
        __syncthreads();
    }
    if (tid < C_CLS) nproto_g[tid] = s_n[tid];
}

// ---------------------------------------------------------------------------
// Kernel 2: per-row sum / sum-of-squares for features (8192) and protos (1280)
// one wave per row
// ---------------------------------------------------------------------------
__global__ void row_stats(const float* __restrict__ F, const float* __restrict__ Pm,
                          float* __restrict__ f2, float* __restrict__ sf,
                          float* __restrict__ p2, float* __restrict__ sp) {
    const int row  = blockIdx.x * 8 + (threadIdx.x >> 5);
    const int lane = threadIdx.x & 31;
    if (row >= B_N + CP) return;
    const float* src = (row < B_N) ? (F + (long long)row * D_DIM)
                                   : (Pm + (long long)(row - B_N) * D_DIM);
    float s = 0.0f, ss = 0.0f;
    for (int d = lane; d < D_DIM; d += 32) {
        const float x = src[d];
        s += x; ss += x * x;
    }
    for (int off = 16; off > 0; off >>= 1) {
        s  += __shfl_xor(s,  off, 32);
        ss += __shfl_xor(ss, off, 32);
    }
    if (lane == 0) {
        if (row < B_N) { f2[row] = ss; sf[row] = s; }
        else           { p2[row - B_N] = ss; sp[row - B_N] = s; }
    }
}

// ---------------------------------------------------------------------------
// Kernel 3: LDS-tiled fp32 WMMA GEMM.
//   block tile 64x64, 8 waves in a 4x2 grid, each wave 16x32 (2 accumulators,
//   A fragment reused). K chunked by 32, A/B chunks staged in LDS with
//   coalesced float4 loads. Epilogue: distances + per-(row,class) exp-sum /
//   min-d2 via wave butterflies + atomics. EXEC all ones before WMMA.
// ---------------------------------------------------------------------------
__global__ void gemm_loss_tiles(const float* __restrict__ F, const float* __restrict__ Pm,
                                const float* __restrict__ f2, const float* __restrict__ sf,
                                const float* __restrict__ p2, const float* __restrict__ sp,
                                const int* __restrict__ nproto,
                                float* __restrict__ pcs, unsigned* __restrict__ dminbits) {
    __shared__ __align__(16) float sA[64 * LDS_STRIDE];   // 64 rows x 32 (+pad)
    __shared__ __align__(16) float sB[64 * LDS_STRIDE];

    const int rt = blockIdx.x;        // 64-row tile
    const int ct = blockIdx.y;        // 64-col tile (within one class: 128%64==0)
    const int tid  = threadIdx.x;
    const int wv   = tid >> 5;        // wave 0..7
    const int lane = tid & 31;
    const int half = lane >> 4;       // fragment K phase: 0 -> K,K+1 ; 1 -> K+2,K+3
    const int lp   = lane & 15;
    const int mw   = wv & 3;          // wave row 0..3  (16 rows each)
    const int nw   = wv >> 2;         // wave col 0..1  (32 cols each)

    const float* aFrag  = &sA[(16 * mw + lp) * LDS_STRIDE];
    const float* bFrag0 = &sB[(32 * nw + lp) * LDS_STRIDE];
    const float* bFrag1 = &sB[(32 * nw + 16 + lp) * LDS_STRIDE];

    v8f acc0 = {}, acc1 = {};

    for (int kb = 0; kb < D_DIM; kb += 32) {
        __syncthreads();
        // stage A (64x32) and B (64x32) chunks: 512 float4 each, 2 per thread
        for (int t = tid; t < 512; t += 256) {
            const int r  = t >> 3;       // 0..63
            const int c4 = (t & 7) * 4;  // 0,4,...,28
            const float4 va = *reinterpret_cast<const float4*>(
                F + (long long)(rt * 64 + r) * D_DIM + kb + c4);
            const float4 vb = *reinterpret_cast<const float4*>(
                Pm + (long long)(ct * 64 + r) * D_DIM + kb + c4);
            *reinterpret_cast<float4*>(&sA[r * LDS_STRIDE + c4]) = va;
            *reinterpret_cast<float4*>(&sB[r * LDS_STRIDE + c4]) = vb;
        }
        __syncthreads();

        #pragma unroll
        for (int kk = 0; kk < 32; kk += 4) {
            const v2f a  = *reinterpret_cast<const v2f*>(aFrag  + kk + 2 * half);
            const v2f b0 = *reinterpret_cast<const v2f*>(bFrag0 + kk + 2 * half);
            const v2f b1 = *reinterpret_cast<const v2f*>(bFrag1 + kk + 2 * half);
            acc0 = __builtin_amdgcn_wmma_f32_16x16x4_f32(false, a, false, b0,
                                                         (short)0, acc0, false, false);
            acc1 = __builtin_amdgcn_wmma_f32_16x16x4_f32(false, a, false, b1,
                                                         (short)0, acc1, false, false);
        }
    }

    // ---- epilogue -----------------------------------------------------------
    const int cls = ct >> 1;                       // 64-col tile within one class
    const int n   = nproto[cls];
    const int cic0 = (ct & 1) * 64 + 32 * nw + lp; // slot index within class, acc0 cols
    const int cic1 = cic0 + 16;                    // acc1 cols
    const bool val0 = cic0 < n;
    const bool val1 = cic1 < n;
    const int col0 = ct * 64 + 32 * nw + lp;
    const int col1 = col0 + 16;
    const float p20 = p2[col0], sp0 = sp[col0];
    const float p21 = p2[col1], sp1 = sp[col1];

    float ev[8], mv[8];
    #pragma unroll
    for (int v = 0; v < 8; ++v) {
        const int row = rt * 64 + 16 * mw + half * 8 + v;
        const float f2v = f2[row], sfv = sf[row];
        float d20 = f2v + p20 - 2.0f * acc0[v]
                  + 2.0f * EPSV * (sfv - sp0) + (float)D_DIM * EPSV * EPSV;
        float d21 = f2v + p21 - 2.0f * acc1[v]
                  + 2.0f * EPSV * (sfv - sp1) + (float)D_DIM * EPSV * EPSV;
        d20 = fmaxf(d20, 0.0f);
        d21 = fmaxf(d21, 0.0f);
        const float e0 = val0 ? __expf(-GAMMAV * d20) : 0.0f;
        const float e1 = val1 ? __expf(-GAMMAV * d21) : 0.0f;
        ev[v] = e0 + e1;
        mv[v] = fminf(val0 ? d20 : __builtin_inff(),
                      val1 ? d21 : __builtin_inff());
    }
    // butterfly over the 16 columns (stays within each 16-lane half)
    #pragma unroll
    for (int off = 1; off < 16; off <<= 1) {
        #pragma unroll
        for (int v = 0; v < 8; ++v) {
            ev[v] += __shfl_xor(ev[v], off, 32);
            mv[v]  = fminf(mv[v], __shfl_xor(mv[v], off, 32));
        }
    }
    if (lp == 0) {
        #pragma unroll
        for (int v = 0; v < 8; ++v) {
            const int row = rt * 64 + 16 * mw + half * 8 + v;
            atomicAdd(&pcs[row * C_CLS + cls], ev[v]);
            atomicMin(&dminbits[row * C_CLS + cls], __float_as_uint(mv[v]));
        }
    }
}

// ---------------------------------------------------------------------------
// Kernel 4: per-sample dce + pairwise terms, reduce to scalar
// ---------------------------------------------------------------------------
__global__ void finalize_loss(const float* __restrict__ pcs, const unsigned* __restrict__ dminbits,
                              const int* __restrict__ labels, const int* __restrict__ nproto,
                              float* __restrict__ out) {
    const int b = blockIdx.x * blockDim.x + threadIdx.x;
    float local = 0.0f;
    if (b < B_N) {
        const int lb = labels[b];
        float one = 0.0f;
        float pc[C_CLS];
        for (int c = 0; c < C_CLS; ++c) { pc[c] = pcs[b * C_CLS + c]; one += pc[c]; }
        const float num  = pc[lb];
        const float prob = 1e-6f + ((one > 0.0f) ? (num / one) : one);
        local = -logf(prob);
        float pw = 0.0f;
        for (int c = 0; c < C_CLS; ++c) {
            if (nproto[c] > 0) {
                const float dmin = sqrtf(__uint_as_float(dminbits[b * C_CLS + c]));
                const float sign = (lb == c) ? 1.0f : -1.0f;
                const float z = 10.0f - (1.0f - dmin) * sign;        // B - (TAO - dmin)*sign
                const float g = (z > 10.0f) ? z : log1pf(__expf(z)); // softplus, beta=1
                pw += g;
            }
        }
        local += 0.1f * pw;   // LAMBDA
    }
    __shared__ float red[256];
    red[threadIdx.x] = local;
    __syncthreads();
    for (int s = 128; s > 0; s >>= 1) {
        if (threadIdx.x < s) red[threadIdx.x] += red[threadIdx.x + s];
        __syncthreads();
    }
    if (threadIdx.x == 0) atomicAdd(out, red[0]);
}

// ---------------------------------------------------------------------------
extern "C" void kernel_launch(void* const* d_in, const int* in_sizes, int n_in,
                              void* d_out, int out_size, void* d_ws, size_t ws_size,
                              hipStream_t stream) {
    const float* F = (const float*)d_in[0];   // [8192, 512] fp32
    const int*   L = (const int*)d_in[1];     // [8192] int32
    float* out = (float*)d_out;               // scalar fp32

    // workspace layout (floats unless noted)
    float* ws       = (float*)d_ws;
    float* protos   = ws;                          // CP*D = 655360
    float* f2       = protos + (size_t)CP * D_DIM; // 8192
    float* sf       = f2 + B_N;                    // 8192
    float* p2       = sf + B_N;                    // 1280
    float* sp       = p2 + CP;                     // 1280
    float* pcs      = sp + CP;                     // B*C = 81920
    unsigned* dminb = (unsigned*)(pcs + (size_t)B_N * C_CLS); // 81920 (uint bits)
    int* nproto     = (int*)(dminb + (size_t)B_N * C_CLS);    // 10

    // 0) init (covers protos span = largest region)
    init_ws<<<(CP * D_DIM + 255) / 256, 256, 0, stream>>>(protos, pcs, dminb, nproto, out);
    // 1) sequential prototype assignment
    assign_protos<<<1, 256, 0, stream>>>(F, L, protos, nproto);
    // 2) row statistics (|x|^2, sum x) for features + prototypes
    row_stats<<<(B_N + CP + 7) / 8, 256, 0, stream>>>(F, protos, f2, sf, p2, sp);
    // 3) LDS-tiled fp32 WMMA GEMM + distance/exp/min epilogue
    dim3 g(B_N / 64, CP / 64);
    gemm_loss_tiles<<<g, 256, 0, stream>>>(F, protos, f2, sf, p2, sp, nproto, pcs, dminb);
    // 4) final scalar reduction
    finalize_loss<<<B_N / 256, 256, 0, stream>>>(pcs, dminb, L, nproto, out);
}